// DynamicNet_58591943852321
// MI455X (gfx1250) — compile-verified
//
#include <hip/hip_runtime.h>
#include <math.h>

// ---------------- problem constants ----------------
#define BATCH   8
#define NNODE   2048
#define NTOT    (BATCH*NNODE)      // 16384
#define KNN     4
#define INCH    16
#define HID     128
#define NCLUS   64

typedef __attribute__((ext_vector_type(2))) float v2f;
typedef __attribute__((ext_vector_type(8))) float v8f;

__device__ __forceinline__ v8f wmma4(v2f a, v2f b, v8f c) {
    // D = A(16x4,f32) * B(4x16,f32) + C(16x16,f32)
    return __builtin_amdgcn_wmma_f32_16x16x4_f32(false, a, false, b,
                                                 (short)0, c, false, false);
}

// ---------------- utility kernels ----------------
__global__ void fill_f32(float* __restrict__ p, int n, float v) {
    int i = blockIdx.x * blockDim.x + threadIdx.x;
    int stride = gridDim.x * blockDim.x;
    for (; i < n; i += stride) p[i] = v;
}

__global__ void count_deg(const int* __restrict__ dst, float* __restrict__ deg, int E) {
    int e = blockIdx.x * blockDim.x + threadIdx.x;
    if (e < E) atomicAdd(&deg[dst[e]], 1.0f);
}

__global__ void rsqrt_k(const float* __restrict__ deg, float* __restrict__ dis, int n) {
    int i = blockIdx.x * blockDim.x + threadIdx.x;
    if (i < n) dis[i] = rsqrtf(deg[i]);
}

// ---------------- WMMA GEMM: Out[MxN] = A[MxK] * B[KxN] --------------------
// B staged TRANSPOSED in dynamic LDS with pad stride (K+4):
//   - b-fragment is one conflict-free ds_load_b64
//   - a-fragment is one global_load_b64
// grid.x = M/16, blockDim = 32 * (N/16)  (N/16 <= 8), shmem = N*(K+4)*4 bytes
__global__ void gemm_bsmall(const float* __restrict__ A, const float* __restrict__ Bw,
                            float* __restrict__ Out, int M, int N, int K) {
    extern __shared__ float lbT[];          // [N][K+4]
    const int KP = K + 4;
    int tid = threadIdx.x;
    int lgN = 31 - __clz(N);                // N is a power of two (64 or 128)
    for (int i = tid; i < K * N; i += blockDim.x) {
        int k = i >> lgN;
        int n = i & (N - 1);
        lbT[n * KP + k] = Bw[i];
    }
    __syncthreads();

    int wave = tid >> 5;
    int lane = tid & 31;
    int half = lane >> 4, lm = lane & 15;
    int m0 = blockIdx.x * 16;
    int n0 = wave * 16;

    const float* arow = A + (size_t)(m0 + lm) * K + 2 * half;
    const float* brow = lbT + (n0 + lm) * KP + 2 * half;

    v8f c = {};
    for (int kk = 0; kk < K; kk += 4) {
        v2f a = *(const v2f*)(arow + kk);
        v2f b = *(const v2f*)(brow + kk);
        c = wmma4(a, b, c);
    }
#pragma unroll
    for (int r = 0; r < 8; ++r)
        Out[(size_t)(m0 + r + 8 * half) * N + n0 + lm] = c[r];
}

// ---------------- conv1 scatter: acc[dst] += xw1[src]*dis[src] -------------
__global__ void scatter_conv1(const float* __restrict__ xw1, const int* __restrict__ src,
                              const int* __restrict__ dst, const float* __restrict__ dis,
                              float* __restrict__ acc) {
    int e = blockIdx.x;
    int c = threadIdx.x;                 // 128
    int si = src[e], di = dst[e];
    float v = xw1[(size_t)si * HID + c] * dis[si];
    atomicAdd(&acc[(size_t)di * HID + c], v);
}

__global__ void finalize1(const float* __restrict__ xw1, const float* __restrict__ acc,
                          const float* __restrict__ dis, const float* __restrict__ b1,
                          float* __restrict__ h1) {
    int idx = blockIdx.x * blockDim.x + threadIdx.x;
    if (idx >= NTOT * HID) return;
    int n = idx >> 7, c = idx & 127;
    float dn = dis[n];
    float v = dn * (acc[idx] + xw1[idx] * dn) + b1[c];
    h1[idx] = fmaxf(v, 0.0f);
}

// ---------------- kNN (k=4) on h1[:, :3], coords in LDS --------------------
__global__ void knn_kernel(const float* __restrict__ h1, int* __restrict__ src2) {
    __shared__ float cx[NNODE], cy[NNODE], cz[NNODE];
    int g = blockIdx.x >> 3;                 // 8 blocks per graph
    int base = (blockIdx.x & 7) * 256;
    const float* hg = h1 + (size_t)g * NNODE * HID;
    for (int i = threadIdx.x; i < NNODE; i += 256) {
        const float* p = hg + (size_t)i * HID;
        cx[i] = p[0]; cy[i] = p[1]; cz[i] = p[2];
    }
    __syncthreads();
    int n = base + threadIdx.x;
    float px = cx[n], py = cy[n], pz = cz[n];
    float bd[KNN] = {1e30f, 1e30f, 1e30f, 1e30f};
    int   bi[KNN] = {0, 0, 0, 0};
    for (int j = 0; j < NNODE; ++j) {
        if (j == n) continue;
        float dx = cx[j] - px, dy = cy[j] - py, dz = cz[j] - pz;
        float d = dx * dx + dy * dy + dz * dz;
        if (d < bd[KNN - 1]) {
            int k = KNN - 1;
            while (k > 0 && d < bd[k - 1]) { bd[k] = bd[k - 1]; bi[k] = bi[k - 1]; --k; }
            bd[k] = d; bi[k] = j;
        }
    }
    int gn = g * NNODE + n;
#pragma unroll
    for (int t = 0; t < KNN; ++t) src2[gn * KNN + t] = g * NNODE + bi[t];
}

// ---------------- conv2: deg==5 uniformly => norm = 1/5 --------------------
__global__ void finalize2(const float* __restrict__ xw2, const int* __restrict__ src2,
                          const float* __restrict__ b2, float* __restrict__ h2) {
    int idx = blockIdx.x * blockDim.x + threadIdx.x;
    if (idx >= NTOT * HID) return;
    int n = idx >> 7, c = idx & 127;
    float acc = xw2[idx];
    const int* nb = src2 + n * KNN;
#pragma unroll
    for (int t = 0; t < KNN; ++t) acc += xw2[(size_t)nb[t] * HID + c];
    h2[idx] = fmaxf(0.2f * acc + b2[c], 0.0f);
}

// ---------------- softmax over 64 logits per row (adds bias) ---------------
__global__ void softmax64(float* __restrict__ s, const float* __restrict__ bp) {
    int row = blockIdx.x * 8 + (threadIdx.x >> 5);
    int lane = threadIdx.x & 31;
    float* p = s + (size_t)row * NCLUS;
    float z0 = p[lane] + bp[lane];
    float z1 = p[lane + 32] + bp[lane + 32];
    float mx = fmaxf(z0, z1);
    for (int o = 16; o; o >>= 1) mx = fmaxf(mx, __shfl_xor(mx, o, 32));
    float e0 = __expf(z0 - mx), e1 = __expf(z1 - mx);
    float sum = e0 + e1;
    for (int o = 16; o; o >>= 1) sum += __shfl_xor(sum, o, 32);
    float inv = 1.0f / sum;
    p[lane] = e0 * inv;
    p[lane + 32] = e1 * inv;
}

// ---------------- t = adj @ s via edge scatter ------------------------------
__global__ void scatter_t(const float* __restrict__ s, const int* __restrict__ src,
                          const int* __restrict__ dst, float* __restrict__ t) {
    int e = blockIdx.x;
    int c = threadIdx.x;                 // 64
    atomicAdd(&t[(size_t)src[e] * NCLUS + c], s[(size_t)dst[e] * NCLUS + c]);
}

// ---------------- ss[b] = s_b^T s_b  (64x2048x64 per batch, WMMA) ----------
__global__ void wmma_ss(const float* __restrict__ s, float* __restrict__ ss) {
    int wid = (blockIdx.x * blockDim.x + threadIdx.x) >> 5;  // 8*4*4 = 128 waves
    int b = wid >> 4;
    int mt = (wid >> 2) & 3, nt = wid & 3;
    int lane = threadIdx.x & 31;
    int half = lane >> 4, lm = lane & 15;
    const float* S = s + (size_t)b * NNODE * NCLUS;
    int m0 = mt * 16, n0 = nt * 16;
    v8f c = {};
    for (int kk = 0; kk < NNODE; kk += 4) {
        int ka = kk + 2 * half;
        v2f a, bb;
        a.x  = S[(size_t)ka * NCLUS + m0 + lm];
        a.y  = S[(size_t)(ka + 1) * NCLUS + m0 + lm];
        bb.x = S[(size_t)ka * NCLUS + n0 + lm];
        bb.y = S[(size_t)(ka + 1) * NCLUS + n0 + lm];
        c = wmma4(a, bb, c);
    }
    float* O = ss + (size_t)b * NCLUS * NCLUS;
#pragma unroll
    for (int r = 0; r < 8; ++r)
        O[(m0 + r + 8 * half) * NCLUS + n0 + lm] = c[r];
}

// ---------------- out[b] = selu(s_b^T h2_b) (64x2048x128, WMMA) ------------
__global__ void wmma_bcf_selu(const float* __restrict__ s, const float* __restrict__ h2,
                              float* __restrict__ out) {
    int wid = (blockIdx.x * blockDim.x + threadIdx.x) >> 5;  // 8*4*8 = 256 waves
    int b = wid >> 5;
    int mt = (wid >> 3) & 3, nt = wid & 7;
    int lane = threadIdx.x & 31;
    int half = lane >> 4, lm = lane & 15;
    const float* S = s  + (size_t)b * NNODE * NCLUS;
    const float* H = h2 + (size_t)b * NNODE * HID;
    int m0 = mt * 16, n0 = nt * 16;
    v8f c = {};
    for (int kk = 0; kk < NNODE; kk += 4) {
        int ka = kk + 2 * half;
        v2f a, bb;
        a.x  = S[(size_t)ka * NCLUS + m0 + lm];
        a.y  = S[(size_t)(ka + 1) * NCLUS + m0 + lm];
        bb.x = H[(size_t)ka * HID + n0 + lm];
        bb.y = H[(size_t)(ka + 1) * HID + n0 + lm];
        c = wmma4(a, bb, c);
    }
    const float alpha = 1.6732632423543772f, scale = 1.0507009873554805f;
    float* O = out + (size_t)b * NCLUS * HID;
#pragma unroll
    for (int r = 0; r < 8; ++r) {
        float v = c[r];
        v = (v > 0.0f) ? scale * v : scale * alpha * (__expf(v) - 1.0f);
        O[(size_t)(m0 + r + 8 * half) * HID + n0 + lm] = v;
    }
}

// ---------------- log_softmax over 128 cols per row ------------------------
__global__ void logsoftmax128(float* __restrict__ out) {
    int row = blockIdx.x * 8 + (threadIdx.x >> 5);   // 512 rows
    int lane = threadIdx.x & 31;
    float* p = out + (size_t)row * HID;
    float z[4];
    float mx = -1e30f;
#pragma unroll
    for (int t = 0; t < 4; ++t) { z[t] = p[lane + 32 * t]; mx = fmaxf(mx, z[t]); }
    for (int o = 16; o; o >>= 1) mx = fmaxf(mx, __shfl_xor(mx, o, 32));
    float sum = 0.0f;
#pragma unroll
    for (int t = 0; t < 4; ++t) sum += __expf(z[t] - mx);
    for (int o = 16; o; o >>= 1) sum += __shfl_xor(sum, o, 32);
    float lse = mx + logf(sum);
#pragma unroll
    for (int t = 0; t < 4; ++t) p[lane + 32 * t] = z[t] - lse;
}

// ---------------- per-batch m[b], tr[b] ------------------------------------
__global__ void batch_mtr(const float* __restrict__ deg, const float* __restrict__ s,
                          const float* __restrict__ t, float* __restrict__ m,
                          float* __restrict__ tr) {
    int b = blockIdx.x, tid = threadIdx.x;   // 256 threads
    float sm = 0.0f, st = 0.0f;
    for (int n = tid; n < NNODE; n += 256) sm += deg[b * NNODE + n] - 1.0f;
    for (int i = tid; i < NNODE * NCLUS; i += 256) {
        size_t off = (size_t)b * NNODE * NCLUS + i;
        st += s[off] * t[off];
    }
    __shared__ float r1[256], r2[256];
    r1[tid] = sm; r2[tid] = st; __syncthreads();
    for (int w = 128; w; w >>= 1) {
        if (tid < w) { r1[tid] += r1[tid + w]; r2[tid] += r2[tid + w]; }
        __syncthreads();
    }
    if (!tid) { m[b] = r1[0] * 0.5f; tr[b] = r2[0]; }
}

// ---------------- ca[b,c] = sum_n s*deg ; colsum[b,c] = sum_n s ------------
__global__ void ca_colsum(const float* __restrict__ s, const float* __restrict__ deg,
                          float* __restrict__ ca, float* __restrict__ cs) {
    int p = blockIdx.x * 8 + (threadIdx.x >> 5);   // 512 (b,c) pairs
    int lane = threadIdx.x & 31;
    int b = p >> 6, c = p & 63;
    float a = 0.0f, q = 0.0f;
    for (int n = lane; n < NNODE; n += 32) {
        float sv = s[((size_t)b * NNODE + n) * NCLUS + c];
        a += sv * (deg[b * NNODE + n] - 1.0f);
        q += sv;
    }
    for (int o = 16; o; o >>= 1) { a += __shfl_xor(a, o, 32); q += __shfl_xor(q, o, 32); }
    if (!lane) { ca[p] = a; cs[p] = q; }
}

// ---------------- final scalar: spectral + ortho + cluster -----------------
__device__ __forceinline__ float block_reduce256(float v, float* red, int tid) {
    red[tid] = v; __syncthreads();
    for (int w = 128; w; w >>= 1) {
        if (tid < w) red[tid] += red[tid + w];
        __syncthreads();
    }
    float r = red[0]; __syncthreads();
    return r;
}

__global__ void final_loss(const float* __restrict__ m, const float* __restrict__ tr,
                           const float* __restrict__ ca, const float* __restrict__ cs,
                           const float* __restrict__ ss, float* __restrict__ out_loss) {
    __shared__ float red[256];
    int tid = threadIdx.x;
    float spectral = 0.0f, ortho = 0.0f, cluster = 0.0f;
    for (int b = 0; b < BATCH; ++b) {
        float v = (tid < NCLUS) ? ca[b * NCLUS + tid] * ca[b * NCLUS + tid] : 0.0f;
        float caSq = block_reduce256(v, red, tid);
        v = (tid < NCLUS) ? cs[b * NCLUS + tid] * cs[b * NCLUS + tid] : 0.0f;
        float csSq = block_reduce256(v, red, tid);
        const float* S = ss + (size_t)b * NCLUS * NCLUS;
        v = 0.0f;
        for (int i = tid; i < NCLUS * NCLUS; i += 256) v += S[i] * S[i];
        float fro = sqrtf(block_reduce256(v, red, tid));
        v = 0.0f;
        for (int i = tid; i < NCLUS * NCLUS; i += 256) {
            float e = S[i] / fro - ((i % (NCLUS + 1)) == 0 ? 0.125f : 0.0f);
            v += e * e;
        }
        float orthoSq = block_reduce256(v, red, tid);
        float mb = m[b];
        spectral += (-tr[b] + caSq / (2.0f * mb)) / (2.0f * mb);
        ortho    += sqrtf(orthoSq);
        cluster  += sqrtf(csSq) / (float)NNODE * 8.0f - 1.0f;
    }
    if (!tid) out_loss[0] = (spectral + ortho + cluster) * (1.0f / (float)BATCH);
}

// ======================= host launcher =====================================
extern "C" void kernel_launch(void* const* d_in, const int* in_sizes, int n_in,
                              void* d_out, int out_size, void* d_ws, size_t ws_size,
                              hipStream_t stream) {
    (void)n_in; (void)out_size; (void)ws_size;
    const float* x   = (const float*)d_in[0];
    const int* esrc  = (const int*)d_in[1];
    const int* edst  = (const int*)d_in[2];
    // d_in[3] = batch (unused: equal-size contiguous graphs)
    const float* W1  = (const float*)d_in[4];
    const float* b1  = (const float*)d_in[5];
    const float* W2  = (const float*)d_in[6];
    const float* b2  = (const float*)d_in[7];
    const float* Wp  = (const float*)d_in[8];
    const float* bp  = (const float*)d_in[9];
    const int E = in_sizes[1];

    // outputs: [log_softmax(out) : 8*64*128][loss : 1][s : 8*2048*64]
    float* out0  = (float*)d_out;                     // 65536
    float* lossP = out0 + BATCH * NCLUS * HID;        // 1
    float* sP    = lossP + 1;                         // 1048576

    // workspace layout (floats)
    float* ws   = (float*)d_ws;
    float* bufA = ws;                       // xw1 then xw2 (2097152)
    float* bufB = ws + 2097152;             // acc1 then h2  (2097152)
    float* h1   = ws + 4194304;             // 2097152
    float* tbuf = ws + 6291456;             // 1048576
    float* ssb  = ws + 7340032;             // 32768
    float* deg  = ws + 7372800;             // 16384
    float* dis  = ws + 7389184;             // 16384
    float* mB   = ws + 7405568;             // 8
    float* trB  = ws + 7405576;             // 8
    float* caB  = ws + 7405584;             // 512
    float* csB  = ws + 7406096;             // 512
    int*   src2 = (int*)(ws + 7406608);     // 65536 ints

    // dynamic LDS bytes for the three GEMM shapes: N*(K+4)*4
    const size_t shm1 = (size_t)HID   * (INCH + 4) * sizeof(float); // x@W1
    const size_t shm2 = (size_t)HID   * (HID  + 4) * sizeof(float); // h@W2 (66KB)
    const size_t shmP = (size_t)NCLUS * (HID  + 4) * sizeof(float); // h@Wp

    // ---- init (fresh every call; graph-replay safe) ----
    fill_f32<<<2048, 256, 0, stream>>>(bufB, NTOT * HID, 0.0f);     // acc1 = 0
    fill_f32<<<64,   256, 0, stream>>>(deg,  NTOT, 1.0f);           // self-loop
    fill_f32<<<1024, 256, 0, stream>>>(tbuf, NTOT * NCLUS, 0.0f);   // t = 0

    // ---- conv1 ----
    count_deg<<<(E + 255) / 256, 256, 0, stream>>>(edst, deg, E);
    rsqrt_k<<<(NTOT + 255) / 256, 256, 0, stream>>>(deg, dis, NTOT);
    gemm_bsmall<<<NTOT / 16, 256, shm1, stream>>>(x, W1, bufA, NTOT, HID, INCH);   // xw1
    scatter_conv1<<<E, HID, 0, stream>>>(bufA, esrc, edst, dis, bufB);
    finalize1<<<NTOT * HID / 256, 256, 0, stream>>>(bufA, bufB, dis, b1, h1);

    // ---- dynamic kNN graph on h1[:, :3] ----
    knn_kernel<<<BATCH * (NNODE / 256), 256, 0, stream>>>(h1, src2);

    // ---- conv2 (deg == 5 uniformly => norm = 1/5) ----
    gemm_bsmall<<<NTOT / 16, 256, shm2, stream>>>(h1, W2, bufA, NTOT, HID, HID);   // xw2
    finalize2<<<NTOT * HID / 256, 256, 0, stream>>>(bufA, src2, b2, bufB);         // h2

    // ---- cluster assignments s = softmax(h2 @ Wp + bp) ----
    gemm_bsmall<<<NTOT / 16, 128, shmP, stream>>>(bufB, Wp, sP, NTOT, NCLUS, HID); // logits
    softmax64<<<NTOT / 8, 256, 0, stream>>>(sP, bp);

    // ---- DMoN pooling pieces ----
    scatter_t<<<E, NCLUS, 0, stream>>>(sP, esrc, edst, tbuf);                   // t = adj @ s
    wmma_ss<<<16, 256, 0, stream>>>(sP, ssb);                                   // s^T s
    wmma_bcf_selu<<<32, 256, 0, stream>>>(sP, bufB, out0);                      // selu(s^T h2)
    logsoftmax128<<<BATCH * NCLUS / 8, 256, 0, stream>>>(out0);

    // ---- losses ----
    batch_mtr<<<BATCH, 256, 0, stream>>>(deg, sP, tbuf, mB, trB);
    ca_colsum<<<BATCH * NCLUS / 8, 256, 0, stream>>>(sP, deg, caB, csB);
    final_loss<<<1, 256, 0, stream>>>(mB, trB, caB, csB, ssb, lossP);
}